// PhraseClassifier_57380763074838
// MI455X (gfx1250) — compile-verified
//
#include <hip/hip_runtime.h>
#include <hip/hip_bf16.h>

// ---- problem constants (match reference) ----
#define SEQ_LEN 2048
#define BATCH   16
#define HID     512
#define NPOS    (SEQ_LEN * BATCH)     // 32768 (t,b) positions
#define FEAT    (2 * HID)             // 1024 floats per position
#define EPS_F   1e-12f

typedef __attribute__((ext_vector_type(2))) float v2f;
typedef __attribute__((ext_vector_type(8))) float v8f;

// -----------------------------------------------------------------------------
// Kernel 1: precompute fwdDot[pos] = hidden[pos, 0:512]·W[0:512]
//                      bwdDot[pos] = hidden[pos, 512:1024]·W[512:1024]
// One wave (32 lanes) per block, 16 rows (positions) per wave, via
// V_WMMA_F32_16X16X4_F32.  B carries W in column 0, zeros elsewhere.
// A-frag layout (ISA 7.12.2, 32-bit A 16x4): lane l, M = l&15,
//   v0 = A[M][k0 + koff], v1 = A[M][k0 + koff + 1], koff = (l>>4)*2.
// B-frag layout (rows striped across lanes, 4x16): VGPR0 holds K rows {0,2},
//   VGPR1 holds K rows {1,3} -> same koff as A; column index n = l&15.
// D layout: VGPR r, lanes 0-15 -> M=r, lanes 16-31 -> M=r+8, N = l&15.
//
// W is loaded UNCONDITIONALLY in all lanes (uniform broadcast address) and
// masked to column 0 by a pure-arithmetic multiply -> no exec-mask branches
// in the hot loop (previous version predicated the W loads with
// s_and_saveexec per iteration).
// -----------------------------------------------------------------------------
__global__ __launch_bounds__(32)
void phrase_dot_wmma_kernel(const float* __restrict__ hidden,
                            const float* __restrict__ W,
                            float* __restrict__ fwdDot,
                            float* __restrict__ bwdDot) {
    const int pos0 = blockIdx.x * 16;
    const int lane = threadIdx.x;       // 0..31 (wave32)
    const int m    = lane & 15;         // row within tile / B column index
    const int koff = (lane >> 4) * 2;   // 0 for lanes 0-15, 2 for lanes 16-31

    const float* __restrict__ rowA = hidden + (size_t)(pos0 + m) * FEAT;
    const float colmask = (m == 0) ? 1.0f : 0.0f;   // keep only B column 0

    v8f cf = {};  // fwd accumulator (16x16 f32 D)
    v8f cb = {};  // bwd accumulator

    #pragma unroll 4
    for (int k0 = 0; k0 < HID; k0 += 4) {
        // Unconditional loads; lanes 0-15 share one address, 16-31 another.
        const float w0 = W[k0 + koff];
        const float w1 = W[k0 + koff + 1];
        const float w2 = W[HID + k0 + koff];
        const float w3 = W[HID + k0 + koff + 1];

        // ---- fwd half: A = hidden[:, k0..k0+3], B col0 = W[k0..k0+3] ----
        v2f a, b;
        a.x = rowA[k0 + koff];
        a.y = rowA[k0 + koff + 1];
        b.x = w0 * colmask;
        b.y = w1 * colmask;
        cf = __builtin_amdgcn_wmma_f32_16x16x4_f32(
                false, a, false, b, (short)0, cf, false, false);

        // ---- bwd half: A = hidden[:, 512+k0..], B col0 = W[512+k0..] ----
        v2f a2, b2;
        a2.x = rowA[HID + k0 + koff];
        a2.y = rowA[HID + k0 + koff + 1];
        b2.x = w2 * colmask;
        b2.y = w3 * colmask;
        cb = __builtin_amdgcn_wmma_f32_16x16x4_f32(
                false, a2, false, b2, (short)0, cb, false, false);
    }

    // Column N=0 of D lives in lanes 0 (M=0..7) and 16 (M=8..15).
    if (m == 0) {
        const int rbase = pos0 + (lane >> 4) * 8;
        #pragma unroll
        for (int r = 0; r < 8; ++r) {
            fwdDot[rbase + r] = cf[r];
            bwdDot[rbase + r] = cb[r];
        }
    }
}

// -----------------------------------------------------------------------------
// Kernel 0: zero the double accumulator (d_ws is poisoned by the harness).
// -----------------------------------------------------------------------------
__global__ void phrase_init_kernel(double* __restrict__ acc) {
    if (threadIdx.x == 0 && blockIdx.x == 0) acc[0] = 0.0;
}

// -----------------------------------------------------------------------------
// Kernel 2: gather 2 precomputed dots per span, sigmoid + BCE, block-reduce,
// global double atomic add.
// -----------------------------------------------------------------------------
__global__ __launch_bounds__(256)
void phrase_bce_kernel(const int* __restrict__ bids,
                       const int* __restrict__ begins,
                       const int* __restrict__ ends,
                       const float* __restrict__ targets,
                       const float* __restrict__ bptr,
                       const float* __restrict__ fwdDot,
                       const float* __restrict__ bwdDot,
                       double* __restrict__ acc,
                       int nspans) {
    __shared__ double sdata[256];
    const int tid = threadIdx.x;
    const float bb = bptr[0];

    double local = 0.0;
    for (int s = blockIdx.x * blockDim.x + tid; s < nspans;
         s += gridDim.x * blockDim.x) {
        const int bid = bids[s];
        const int beg = begins[s];   // in [1, SEQ_LEN)
        const int end = ends[s];     // in [0, SEQ_LEN)
        const float t = targets[s];

        const float x = fwdDot[(beg - 1) * BATCH + bid]
                      + bwdDot[end * BATCH + bid] + bb;

        const float p   = 1.0f / (1.0f + expf(-x));
        const float pc  = fminf(fmaxf(p,        EPS_F), 1.0f);
        const float qc  = fminf(fmaxf(1.0f - p, EPS_F), 1.0f);
        const float bce = -(t * logf(pc) + (1.0f - t) * logf(qc));
        local += (double)bce;
    }

    sdata[tid] = local;
    __syncthreads();
    #pragma unroll
    for (int off = 128; off > 0; off >>= 1) {
        if (tid < off) sdata[tid] += sdata[tid + off];
        __syncthreads();
    }
    if (tid == 0) atomicAdd(acc, sdata[0]);
}

// -----------------------------------------------------------------------------
// Kernel 3: mean = acc / N
// -----------------------------------------------------------------------------
__global__ void phrase_final_kernel(const double* __restrict__ acc,
                                    float* __restrict__ out, int nspans) {
    if (threadIdx.x == 0 && blockIdx.x == 0)
        out[0] = (float)(acc[0] / (double)nspans);
}

extern "C" void kernel_launch(void* const* d_in, const int* in_sizes, int n_in,
                              void* d_out, int out_size, void* d_ws, size_t ws_size,
                              hipStream_t stream) {
    // Inputs in setup_inputs() order:
    const float* hidden  = (const float*)d_in[0];  // (2048, 16, 1024) f32
    const int*   bids    = (const int*)  d_in[1];  // (N,)
    const int*   begins  = (const int*)  d_in[2];  // (N,)
    const int*   ends    = (const int*)  d_in[3];  // (N,)
    const float* targets = (const float*)d_in[4];  // (N,) f32
    const float* W       = (const float*)d_in[5];  // (1024, 1) f32
    const float* bptr    = (const float*)d_in[6];  // (1,)   f32
    const int nspans     = in_sizes[1];

    // Workspace layout: fwdDot[NPOS] | bwdDot[NPOS] | acc (double)
    float*  fwdDot = (float*)d_ws;
    float*  bwdDot = fwdDot + NPOS;
    double* acc    = (double*)(bwdDot + NPOS);     // offset 256 KiB, 8B aligned

    float* out = (float*)d_out;

    phrase_init_kernel<<<1, 32, 0, stream>>>(acc);

    // 32768 positions / 16 rows per wave = 2048 single-wave blocks
    phrase_dot_wmma_kernel<<<NPOS / 16, 32, 0, stream>>>(hidden, W, fwdDot, bwdDot);

    phrase_bce_kernel<<<512, 256, 0, stream>>>(bids, begins, ends, targets, bptr,
                                               fwdDot, bwdDot, acc, nspans);

    phrase_final_kernel<<<1, 32, 0, stream>>>(acc, out, nspans);
}